// CTCBeamSearchDecoder_54434415509589
// MI455X (gfx1250) — compile-verified
//
#include <hip/hip_runtime.h>
#include <math.h>
#include <stdint.h>

#define T_     100
#define V_     96
#define VN     95                 // non-blank symbols
#define W_     8                  // beam width
#define NC     (W_ + W_ * VN)     // 768 candidates
#define NCHUNK 24                 // candidates per lane (768/32)
#define NEG_INF_F (-1.0e30f)
#define USED_F    (-3.0e38f)
#define PREF_W 26                 // 104 bytes per prefix, word-addressed
#define WPB    8                  // waves per block (256 threads, wave32)

// Wave-private LDS slab: no cross-wave sharing -> no s_barrier ever needed.
struct WaveShm {
  float lp[2][V_];            // double-buffered raw logits row (filled by async DMA)
  float cand[NC];             // [0..7]=stay tot, [8..767]=ext_lpnb
  float contrib[W_][W_];      // [w2][w1] merge contributions
  float totS[W_];             // logaddexp(lpb,lpnb)
  float slpb[W_];             // stay_lpb
  float snb[W_];              // final stay_lpnb (after contrib)
  float lpbS[2][W_];          // double-buffered beam state
  float lpnbS[2][W_];
  int   lenS[2][W_];
  int   lastS[2][W_];
  int   topiA[W_];            // selected candidate indices
  float topvA[W_];            // selected candidate totals
  unsigned int pref[2][W_][PREF_W];  // double-buffered prefixes (bytes, word access)
};

// Wave-internal LDS fence: LDS is in-order per wave; this both forces DS
// completion and acts as a compiler reordering barrier.
#define LDS_FENCE() asm volatile("s_wait_dscnt 0x0" ::: "memory")

__device__ __forceinline__ float lax(float a, float b) {  // logaddexp
  float mx = fmaxf(a, b);
  float mn = fminf(a, b);
  return mx + log1pf(expf(mn - mx));
}

// Async global->LDS copy of one 96-float logits row (3 x 32-lane b32 xfers).
// Same immediate offset applies to both the LDS and global side per the ISA.
__device__ __forceinline__ void async_row(unsigned int lds_base, const float* gbase, int lane) {
  unsigned long long ga = (unsigned long long)(uintptr_t)gbase + (unsigned)(lane * 4);
  unsigned int la = lds_base + (unsigned)(lane * 4);
  asm volatile(
      "global_load_async_to_lds_b32 %0, %1, off\n\t"
      "global_load_async_to_lds_b32 %0, %1, off offset:128\n\t"
      "global_load_async_to_lds_b32 %0, %1, off offset:256"
      :: "v"(la), "v"(ga) : "memory");
}

__global__ __launch_bounds__(256)
void ctc_beam_kernel(const float* __restrict__ in, float* __restrict__ out, int B) {
  __shared__ WaveShm shm[WPB];
  const int lane = threadIdx.x & 31;
  const int wv   = threadIdx.x >> 5;
  const int b    = blockIdx.x * WPB + wv;
  if (b >= B) return;
  WaveShm& S = shm[wv];
  const float* row0 = in + (size_t)b * (T_ * V_);

  // Low 32 bits of a generic pointer to __shared__ == group-relative LDS offset.
  const unsigned int lpoff0 = (unsigned int)(uintptr_t)(&S.lp[0][0]);
  const unsigned int lpoff1 = (unsigned int)(uintptr_t)(&S.lp[1][0]);

  if (lane < W_) {
    S.lpbS[0][lane]  = (lane == 0) ? 0.0f : NEG_INF_F;
    S.lpnbS[0][lane] = NEG_INF_F;
    S.lenS[0][lane]  = 0;
    S.lastS[0][lane] = -1;
  }
  LDS_FENCE();

  async_row(lpoff0, row0, lane);   // prefetch t=0

  for (int t = 0; t < T_; ++t) {
    const int buf = t & 1;        // lp buffer & state buffer for step t
    const int nb  = buf ^ 1;      // next lp buffer & new state buffer
    const int cur = buf;

    // ---- 1. prefetch row t+1 (dummy re-load of row t at the tail) --------
    const int tn = (t + 1 < T_) ? (t + 1) : t;
    async_row(nb ? lpoff1 : lpoff0, row0 + (size_t)tn * V_, lane);
    // ---- 2. wait until row t's 3 transfers have landed (<=3 newest pend) --
    asm volatile("s_wait_asynccnt 0x3" ::: "memory");

    // ---- 3. log-softmax of row t (3 values per lane, butterfly reduce) ---
    float x0 = S.lp[buf][lane];
    float x1 = S.lp[buf][lane + 32];
    float x2 = S.lp[buf][lane + 64];
    float m = fmaxf(x0, fmaxf(x1, x2));
    #pragma unroll
    for (int d = 16; d > 0; d >>= 1) m = fmaxf(m, __shfl_xor(m, d, 32));
    float sm = expf(x0 - m) + expf(x1 - m) + expf(x2 - m);
    #pragma unroll
    for (int d = 16; d > 0; d >>= 1) sm += __shfl_xor(sm, d, 32);
    const float lse = m + logf(sm);
    const float lpblank = S.lp[buf][V_ - 1] - lse;

    // ---- 4. per-beam totals / stay_lpb ------------------------------------
    if (lane < W_) {
      float tt = lax(S.lpbS[cur][lane], S.lpnbS[cur][lane]);
      S.totS[lane] = tt;
      S.slpb[lane] = tt + lpblank;
    }
    LDS_FENCE();

    // ---- 5. extend candidates: ext[w][s] = (s==last ? lpb : tot) + lp[s] --
    const float l0 = x0 - lse, l1 = x1 - lse, l2 = x2 - lse;
    #pragma unroll
    for (int w = 0; w < W_; ++w) {
      const float tt = S.totS[w];
      const float lb = S.lpbS[cur][w];
      const int   la = S.lastS[cur][w];
      int s0 = lane;
      S.cand[W_ + w * VN + s0] = ((s0 == la) ? lb : tt) + l0;
      int s1 = lane + 32;
      S.cand[W_ + w * VN + s1] = ((s1 == la) ? lb : tt) + l1;
      int s2 = lane + 64;
      if (s2 < VN) S.cand[W_ + w * VN + s2] = ((s2 == la) ? lb : tt) + l2;
    }
    LDS_FENCE();

    // ---- 6. exact prefix merge: 64 (w1,w2) pairs, 2 per lane --------------
    int msk_s[2], msk_w1[2];
    #pragma unroll
    for (int pp = 0; pp < 2; ++pp) {
      const int p  = lane + 32 * pp;
      const int w1 = p >> 3;
      const int w2 = p & 7;
      float cval = NEG_INF_F;
      int   cs = -1;
      const int L1 = S.lenS[cur][w1];
      if (S.lenS[cur][w2] == L1 + 1) {
        const unsigned int* pa = S.pref[cur][w1];
        const unsigned int* pb = S.pref[cur][w2];
        bool eq = true;
        const int nwords = L1 >> 2;
        for (int k = 0; k < nwords; ++k) eq = eq && (pa[k] == pb[k]);
        const int rem = L1 & 3;
        if (rem) {
          unsigned int mk = (1u << (8 * rem)) - 1u;
          eq = eq && (((pa[nwords] ^ pb[nwords]) & mk) == 0u);
        }
        if (eq) {
          cs   = (int)((pb[L1 >> 2] >> (8 * (L1 & 3))) & 0xFFu);  // pref_w2[L1]
          cval = S.cand[W_ + w1 * VN + cs];                        // pre-mask ext
        }
      }
      S.contrib[w2][w1] = cval;
      msk_s[pp] = cs; msk_w1[pp] = w1;
    }
    LDS_FENCE();

    // ---- 7. mask merged extends; fold contrib into stay_lpnb --------------
    #pragma unroll
    for (int pp = 0; pp < 2; ++pp)
      if (msk_s[pp] >= 0) S.cand[W_ + msk_w1[pp] * VN + msk_s[pp]] = NEG_INF_F;
    if (lane < W_) {
      const int w2 = lane;
      float c = NEG_INF_F;
      #pragma unroll
      for (int w1 = 0; w1 < W_; ++w1) c = lax(c, S.contrib[w2][w1]);
      const int la = S.lastS[cur][w2];
      const float pre = (la >= 0) ? (S.lpnbS[cur][w2] + (S.lp[buf][la] - lse)) : NEG_INF_F;
      const float snbv = lax(pre, c);
      S.snb[w2]  = snbv;
      S.cand[w2] = lax(S.slpb[w2], snbv);   // stay candidate total
    }
    LDS_FENCE();

    // ---- 8. stable top-8: one bulk LDS read, then register-only rounds ----
    float cv[NCHUNK];
    #pragma unroll
    for (int j = 0; j < NCHUNK; ++j) cv[j] = S.cand[lane + 32 * j];
    unsigned int used = 0;   // per-lane mask of already-extracted slots
    #pragma unroll
    for (int r = 0; r < W_; ++r) {
      float bv = -3.4e38f; int bi = NC;
      #pragma unroll
      for (int j = 0; j < NCHUNK; ++j) {
        // tie -> lower index: strict '>' keeps the earliest (j ascending)
        if (!((used >> j) & 1u) && cv[j] > bv) { bv = cv[j]; bi = lane + 32 * j; }
      }
      #pragma unroll
      for (int d = 16; d > 0; d >>= 1) {
        float ov = __shfl_xor(bv, d, 32);
        int   oi = __shfl_xor(bi, d, 32);
        if (ov > bv || (ov == bv && oi < bi)) { bv = ov; bi = oi; }
      }
      if (lane == 0) { S.topiA[r] = bi; S.topvA[r] = bv; }
      if ((bi & 31) == lane) used |= 1u << (bi >> 5);   // owning lane retires slot
    }
    LDS_FENCE();

    // ---- 9. new scalar beam state -----------------------------------------
    if (lane < W_) {
      const int i = S.topiA[lane];
      if (i < W_) {                               // stay
        S.lpbS[nb][lane]  = S.slpb[i];
        S.lpnbS[nb][lane] = S.snb[i];
        S.lenS[nb][lane]  = S.lenS[cur][i];
        S.lastS[nb][lane] = S.lastS[cur][i];
      } else {                                    // extend
        const int e = i - W_;
        const int p = e / VN;
        const int s = e - p * VN;
        S.lpbS[nb][lane]  = NEG_INF_F;
        S.lpnbS[nb][lane] = S.topvA[lane];
        S.lenS[nb][lane]  = S.lenS[cur][p] + 1;
        S.lastS[nb][lane] = s;
      }
    }
    // ---- 10. prefix copy (+ append), 4 lanes per beam ---------------------
    {
      const int r   = lane >> 2;
      const int i   = S.topiA[r];
      const int par = (i < W_) ? i : (i - W_) / VN;
      unsigned int* dst       = S.pref[nb][r];
      const unsigned int* src = S.pref[cur][par];
      #pragma unroll
      for (int k = (lane & 3); k < PREF_W; k += 4) dst[k] = src[k];
      LDS_FENCE();   // copy fully lands before the appended byte overwrite
      if ((lane & 3) == 0 && i >= W_) {
        const int s    = (i - W_) - par * VN;
        const int plen = S.lenS[cur][par];
        ((unsigned char*)S.pref[nb][r])[plen] = (unsigned char)s;
      }
    }
    LDS_FENCE();
  }

  // ---- final: top-1 beam, emit ids (pad -1) as floats + exp(logprob) ------
  const int fin = T_ & 1;   // state parity after T steps (start parity 0)
  if (lane < W_) S.cand[lane] = lax(S.lpbS[fin][lane], S.lpnbS[fin][lane]);
  LDS_FENCE();
  float bv = -3.4e38f; int bw = 0;
  #pragma unroll
  for (int w = 0; w < W_; ++w) {
    float v = S.cand[w];
    if (v > bv) { bv = v; bw = w; }
  }
  const int blen = S.lenS[fin][bw];
  const unsigned char* bp = (const unsigned char*)S.pref[fin][bw];
  float* dec = out + (size_t)b * T_;
  for (int pos = lane; pos < T_; pos += 32)
    dec[pos] = (pos < blen) ? (float)bp[pos] : -1.0f;
  if (lane == 0) out[(size_t)B * T_ + b] = expf(bv);
}

extern "C" void kernel_launch(void* const* d_in, const int* in_sizes, int n_in,
                              void* d_out, int out_size, void* d_ws, size_t ws_size,
                              hipStream_t stream) {
  (void)n_in; (void)out_size; (void)d_ws; (void)ws_size;
  const float* in = (const float*)d_in[0];
  float* out = (float*)d_out;
  const int B = in_sizes[0] / (T_ * V_);
  const int blocks = (B + WPB - 1) / WPB;
  hipLaunchKernelGGL(ctc_beam_kernel, dim3(blocks), dim3(256), 0, stream, in, out, B);
}